// DownSample_32538672235162
// MI455X (gfx1250) — compile-verified
//
#include <hip/hip_runtime.h>

typedef float v2f __attribute__((ext_vector_type(2)));
typedef float v8f __attribute__((ext_vector_type(8)));

#define C_IN  128
#define C_OUT 256
#define KNN   16
#define LDS_STRIDE 132   // 128 + 4 pad: bank advance 4/row, 16B-aligned rows
#define LN_EPS 1e-5f

// One workgroup (8 wave32) per query point (grid-stride loop).
// Phase 0: stage W [256x128] into padded LDS (once per block).
// Phase 1: gather 16 neighbor rows, LayerNorm over 128 channels -> h_lds.
// Phase 2: per wave, two 16-wide c_out tiles: C[16x16] += A[16x4]*B[4x16]
//          via v_wmma_f32_16x16x4_f32, 32 K-steps (K=128), A fragment shared.
// Phase 3: max over the 16 neighbor rows directly from the C-fragment layout.
__global__ __launch_bounds__(256)
void downsample_ln_linear_maxpool_wmma(const float* __restrict__ x,
                                       const int*   __restrict__ knn,
                                       const float* __restrict__ norm_w,
                                       const float* __restrict__ norm_b,
                                       const float* __restrict__ W,
                                       float* __restrict__ out,
                                       int m)
{
    __shared__ float W_lds[C_OUT * LDS_STRIDE];   // ~135 KB
    __shared__ float h_lds[KNN * LDS_STRIDE];     // ~8 KB

    const int tid = threadIdx.x;

    // ---- Phase 0: W -> LDS (each thread copies one 128-float row) ----
    {
        const float* src = W + tid * C_IN;
        float* dst = W_lds + tid * LDS_STRIDE;
        #pragma unroll
        for (int c = 0; c < C_IN; c += 4) {
            float4 v = *(const float4*)(src + c);
            *(float4*)(dst + c) = v;
        }
    }
    __syncthreads();

    const int lane = tid & 31;
    const int wave = tid >> 5;   // 0..7
    const int r    = tid >> 4;   // neighbor row 0..15 (phase 1)
    const int j    = tid & 15;   // lane-in-rowgroup (phase 1)

    // LayerNorm affine params for this thread's 8 strided channels
    float nw[8], nb[8];
    #pragma unroll
    for (int i = 0; i < 8; ++i) {
        nw[i] = norm_w[j + 16 * i];
        nb[i] = norm_b[j + 16 * i];
    }

    for (int q = blockIdx.x; q < m; q += gridDim.x) {
        // ---- Phase 1: gather + LayerNorm into h_lds ----
        {
            const int idx = knn[q * KNN + r];
            const float* src = x + (long)idx * C_IN;
            float v[8];
            float s = 0.f, s2 = 0.f;
            #pragma unroll
            for (int i = 0; i < 8; ++i) {
                v[i] = src[j + 16 * i];
                s  += v[i];
                s2 += v[i] * v[i];
            }
            // reduce across the 16 lanes of this row group (stays in half-wave)
            #pragma unroll
            for (int msk = 8; msk >= 1; msk >>= 1) {
                s  += __shfl_xor(s,  msk, 32);
                s2 += __shfl_xor(s2, msk, 32);
            }
            const float mu   = s  * (1.0f / C_IN);
            const float var  = s2 * (1.0f / C_IN) - mu * mu;
            const float rstd = rsqrtf(var + LN_EPS);
            float* dst = h_lds + r * LDS_STRIDE;
            #pragma unroll
            for (int i = 0; i < 8; ++i)
                dst[j + 16 * i] = (v[i] - mu) * rstd * nw[i] + nb[i];
        }
        __syncthreads();

        // ---- Phase 2: WMMA f32 16x16x4, two N-tiles per wave ----
        {
            const int rowA  = lane & 15;          // M index (A) / N or row index (B)
            const int kpair = (lane >> 4) * 2;    // K sub-offset per half-wave
            const float* aBase  = h_lds + rowA * LDS_STRIDE + kpair;
            const float* b0Base = W_lds + (wave * 16 + rowA) * LDS_STRIDE + kpair;
            const float* b1Base = W_lds + ((wave + 8) * 16 + rowA) * LDS_STRIDE + kpair;

            v8f acc0 = {};
            v8f acc1 = {};
            #pragma unroll
            for (int kb = 0; kb < C_IN / 4; ++kb) {
                v2f a  = *(const v2f*)(aBase  + kb * 4);
                v2f b0 = *(const v2f*)(b0Base + kb * 4);
                v2f b1 = *(const v2f*)(b1Base + kb * 4);
                acc0 = __builtin_amdgcn_wmma_f32_16x16x4_f32(
                           false, a, false, b0, (short)0, acc0, false, false);
                acc1 = __builtin_amdgcn_wmma_f32_16x16x4_f32(
                           false, a, false, b1, (short)0, acc1, false, false);
            }

            // ---- Phase 3: max over M (the 16 neighbors) ----
            // C layout: lane L, vgpr v -> M = v + 8*(L>=16), N = L&15
            float m0 = acc0[0], m1 = acc1[0];
            #pragma unroll
            for (int i2 = 1; i2 < 8; ++i2) {
                m0 = fmaxf(m0, acc0[i2]);
                m1 = fmaxf(m1, acc1[i2]);
            }
            m0 = fmaxf(m0, __shfl_xor(m0, 16, 32));
            m1 = fmaxf(m1, __shfl_xor(m1, 16, 32));
            if (lane < 16) {
                out[(long)q * C_OUT + wave * 16 + lane]       = m0;
                out[(long)q * C_OUT + (wave + 8) * 16 + lane] = m1;
            }
        }
        __syncthreads();   // h_lds reused next iteration
    }
}

extern "C" void kernel_launch(void* const* d_in, const int* in_sizes, int n_in,
                              void* d_out, int out_size, void* d_ws, size_t ws_size,
                              hipStream_t stream) {
    // setup_inputs order: p, x, o, n_p, knn_idx, n_o, norm_w, norm_b, W
    const float* x      = (const float*)d_in[1];
    const float* n_p    = (const float*)d_in[3];
    const int*   knn    = (const int*)  d_in[4];
    const void*  n_o    = (const void*) d_in[5];
    const float* norm_w = (const float*)d_in[6];
    const float* norm_b = (const float*)d_in[7];
    const float* W      = (const float*)d_in[8];

    const int m = in_sizes[4] / KNN;         // 50000
    float* out = (float*)d_out;

    int blocks = (m < 2048) ? m : 2048;      // grid-stride; amortizes W->LDS stage
    downsample_ln_linear_maxpool_wmma<<<blocks, 256, 0, stream>>>(
        x, knn, norm_w, norm_b, W, out, m);

    // Passthrough outputs: n_p then n_o, concatenated after the pooled features.
    const size_t off_elems = (size_t)m * C_OUT;
    const size_t np_elems  = (size_t)in_sizes[3];
    hipMemcpyAsync(out + off_elems, n_p, np_elems * sizeof(float),
                   hipMemcpyDeviceToDevice, stream);

    long tail_bytes = ((long)out_size - (long)(off_elems + np_elems)) * (long)sizeof(float);
    if (tail_bytes > 0) {
        if (tail_bytes > 32) tail_bytes = 32;   // n_o source buffer is <= 32 bytes
        hipMemcpyAsync((char*)d_out + (off_elems + np_elems) * sizeof(float), n_o,
                       (size_t)tail_bytes, hipMemcpyDeviceToDevice, stream);
    }
}